// CommNet_42545946034942
// MI455X (gfx1250) — compile-verified
//
#include <hip/hip_runtime.h>

typedef __attribute__((ext_vector_type(2))) float v2f;
typedef __attribute__((ext_vector_type(4))) float v4f;
typedef __attribute__((ext_vector_type(8))) float v8f;

#define A_N 8
#define B_N 64
#define S_N 1024
#define H_N 128
#define BS_N (B_N * S_N)   // 65536 = b*S + s span
#define M_N (A_N * B_N * S_N)

// Block: 256 threads = 8 wave32 waves. Each wave computes a 16(M) x 128(N)
// output tile with V_WMMA_F32_16X16X4_F32. Block covers 128 rows.
// Grid: M / 128 = 4096 blocks.
__global__ __launch_bounds__(256, 1)
void commnet_fused_kernel(const float* __restrict__ obs,
                          const float* __restrict__ rnn,
                          const int*   __restrict__ alive,
                          const float* __restrict__ W,
                          const float* __restrict__ bias,
                          float* __restrict__ out)
{
    // W staged in LDS (64 KB), rotation-swizzled: W[n][k] -> sW[n*128 + ((k+4n)&127)]
    // so 16 lanes reading 16 different rows at the same k hit distinct banks.
    __shared__ float sW[H_N * H_N];

    const int tid  = threadIdx.x;
    const int lane = tid & 31;
    const int wave = tid >> 5;

    // ---- cooperative stage of W into LDS (float4 per thread x16) ----
    #pragma unroll
    for (int i = 0; i < 16; ++i) {
        const int idx4 = tid + i * 256;        // float4 index 0..4095
        const int n    = idx4 >> 5;            // 32 float4 per 128-float row
        const int k4   = (idx4 & 31) << 2;     // k base (multiple of 4)
        v4f v = *(const v4f*)(W + (idx4 << 2));
        const int dk = (k4 + (n << 2)) & 127;  // swizzle keeps float4 contiguous
        *(v4f*)&sW[n * H_N + dk] = v;
    }
    __syncthreads();

    const int tile_base = blockIdx.x * 128 + wave * 16;
    const int nloc = lane & 15;        // N within a 16-wide tile / A-row id
    const int hi   = lane >> 4;        // 0: K={0,1}, 1: K={2,3} half of wave
    const int kb   = hi << 1;

    // ---- per-lane A-row pointer: row r -> rnn_h[s, b, a, :] ----
    const int r  = tile_base + nloc;
    const int a  = r >> 16;            // r / (B*S)
    const int bs = r & (BS_N - 1);     // b*S + s
    const int bb = bs >> 10;
    const int ss = bs & (S_N - 1);
    const float* aptr = rnn + ((size_t)((ss * B_N + bb) * A_N + a) << 7);

    // ---- row scale = alive[r] / max(sum_a alive[a', b, s], 1) ----
    int cnt = 0;
    #pragma unroll
    for (int aa = 0; aa < A_N; ++aa) cnt += alive[aa * BS_N + bs];
    const float scale = (float)alive[r] / (float)(cnt > 0 ? cnt : 1);

    // ---- main GEMM loop: D(16x16) += A(16x4) x B(4x16), K = 128 ----
    v8f acc[8];
    #pragma unroll
    for (int t = 0; t < 8; ++t) acc[t] = (v8f){0.f,0.f,0.f,0.f,0.f,0.f,0.f,0.f};

    #pragma unroll 2
    for (int kk = 0; kk < 32; ++kk) {
        const int k0 = (kk << 2) + kb;               // this lane's K base
        const v2f afrag = *(const v2f*)(aptr + k0);  // A[m=nloc][k0..k0+1]
        #pragma unroll
        for (int t = 0; t < 8; ++t) {
            const int n = t * 16 + nloc;             // B col = W output row
            // B[k][n] = W[n][k]: contiguous float2 in (swizzled) W row n
            const v2f bfrag =
                *(const v2f*)&sW[n * H_N + ((k0 + (n << 2)) & 127)];
            acc[t] = __builtin_amdgcn_wmma_f32_16x16x4_f32(
                false, afrag, false, bfrag, (short)0, acc[t], false, false);
        }
    }

    // ---- epilogue: out = obs + bias + scale * acc ----
    float bcol[8];
    #pragma unroll
    for (int t = 0; t < 8; ++t) bcol[t] = bias[t * 16 + nloc];

    #pragma unroll
    for (int j = 0; j < 8; ++j) {
        // D VGPR j holds rows j (lanes 0-15) and j+8 (lanes 16-31).
        const int   m  = (hi << 3) + j;
        const float sc = __shfl(scale, m, 32);   // lane m holds row tile_base+m
        const size_t rowbase = ((size_t)(tile_base + m)) << 7;
        #pragma unroll
        for (int t = 0; t < 8; ++t) {
            const size_t idx = rowbase + (size_t)(t * 16 + nloc);
            out[idx] = obs[idx] + bcol[t] + sc * acc[t][j];
        }
    }
}

extern "C" void kernel_launch(void* const* d_in, const int* in_sizes, int n_in,
                              void* d_out, int out_size, void* d_ws, size_t ws_size,
                              hipStream_t stream) {
    const float* obs   = (const float*)d_in[0];
    const float* rnn   = (const float*)d_in[1];
    const int*   alive = (const int*)d_in[2];
    const float* W     = (const float*)d_in[3];
    const float* bias  = (const float*)d_in[4];
    float* out = (float*)d_out;

    dim3 grid(M_N / 128);   // 4096 blocks
    dim3 block(256);        // 8 wave32 waves
    commnet_fused_kernel<<<grid, block, 0, stream>>>(obs, rnn, alive, W, bias, out);
}